// MeshPool_85383949844942
// MI455X (gfx1250) — compile-verified
//
#include <hip/hip_runtime.h>
#include <cstdint>

// Problem constants (match the reference).
#define B_      16
#define C_      256
#define F_      30000
#define TARGET_ 15000

#define FTILE   256   // faces per block tile
#define CHSLAB  64    // channels per block (C_ / CHSLAB = 4 slabs)

// ---------------------------------------------------------------------------
// LDS byte-offset of a pointer into a __shared__ array (addrspace(3) cast).
// ---------------------------------------------------------------------------
__device__ __forceinline__ unsigned lds_offset_of(const void* p) {
    return (unsigned)(unsigned long long)(__attribute__((address_space(3))) const void*)p;
}

// ---------------------------------------------------------------------------
// Main scatter-accumulate (defined FIRST so it leads the disassembly).
// Block: 256 threads = one tile of 256 faces, one 64-channel slab, one mesh.
// Group ids for the tile are DMA'd into LDS with the gfx1250 async path
// (global_load_async_to_lds_b32, ASYNCcnt) and reused across all 64 channels.
// fe is streamed with non-temporal loads (single-use; keep L2 for the output).
// Accumulation uses no-return f32 global atomics (L2 atomic units).
// ---------------------------------------------------------------------------
__global__ void __launch_bounds__(FTILE)
mp_scatter(const float* __restrict__ fe, const int* __restrict__ gid,
           float* __restrict__ out) {
    __shared__ int s_g[FTILE];

    const int tid = threadIdx.x;
    const int f   = blockIdx.x * FTILE + tid;
    const int b   = blockIdx.z;
    const int c0  = blockIdx.y * CHSLAB;
    const bool valid = (f < F_);

    if (valid) {
        unsigned           lds = lds_offset_of(&s_g[tid]);
        unsigned long long src = (unsigned long long)(const void*)(gid + (size_t)b * F_ + f);
        // CDNA5 async global->LDS copy (per-lane, EXEC-masked, tracked by ASYNCcnt)
        asm volatile("global_load_async_to_lds_b32 %0, %1, off"
                     :: "v"(lds), "v"(src) : "memory");
    }
    asm volatile("s_wait_asynccnt 0" ::: "memory");
    __syncthreads();
    if (!valid) return;

    const int    g       = s_g[tid];
    const size_t febase  = ((size_t)b * C_ + c0) * F_      + (size_t)f;
    const size_t outbase = ((size_t)b * C_ + c0) * TARGET_ + (size_t)g;

#pragma unroll 8
    for (int cc = 0; cc < CHSLAB; ++cc) {
        float v = __builtin_nontemporal_load(fe + febase + (size_t)cc * F_);
        unsafeAtomicAdd(out + outbase + (size_t)cc * TARGET_, v);  // global_atomic_add_f32
    }
}

// ---------------------------------------------------------------------------
// Per-mesh group occurrence counts: cnt[b*TARGET + g] += 1.
// ---------------------------------------------------------------------------
__global__ void mp_counts(const int* __restrict__ gid, unsigned* __restrict__ cnt) {
    int i = blockIdx.x * blockDim.x + threadIdx.x;
    if (i < B_ * F_) {
        int b = i / F_;
        int g = gid[i];
        atomicAdd(&cnt[(size_t)b * TARGET_ + g], 1u);   // global_atomic_add_u32, no return
    }
}

// ---------------------------------------------------------------------------
// inv[i] = 1 / max(cnt[i], 1)  (multiply later instead of divide).
// ---------------------------------------------------------------------------
__global__ void mp_recip(const unsigned* __restrict__ cnt, float* __restrict__ inv) {
    int i = blockIdx.x * blockDim.x + threadIdx.x;
    if (i < B_ * TARGET_) {
        unsigned c = cnt[i];
        inv[i] = 1.0f / (float)(c ? c : 1u);
    }
}

// ---------------------------------------------------------------------------
// out[b,c,t] *= inv[b,t]
// ---------------------------------------------------------------------------
__global__ void mp_scale(float* __restrict__ out, const float* __restrict__ inv) {
    int t = blockIdx.x * blockDim.x + threadIdx.x;
    int c = blockIdx.y;
    int b = blockIdx.z;
    if (t < TARGET_) {
        out[((size_t)b * C_ + c) * TARGET_ + t] *= inv[(size_t)b * TARGET_ + t];
    }
}

// ---------------------------------------------------------------------------
// Zero-fill (float4 vectorized). n4 = number of float4 elements.
// ---------------------------------------------------------------------------
__global__ void mp_zero_f4(float4* __restrict__ p, long long n4) {
    long long i = (long long)blockIdx.x * blockDim.x + threadIdx.x;
    if (i < n4) p[i] = make_float4(0.f, 0.f, 0.f, 0.f);
}

// ---------------------------------------------------------------------------
extern "C" void kernel_launch(void* const* d_in, const int* in_sizes, int n_in,
                              void* d_out, int out_size, void* d_ws, size_t ws_size,
                              hipStream_t stream) {
    const float* fe  = (const float*)d_in[0];   // [B, C, F] f32
    const int*   gid = (const int*)d_in[1];     // [B, F] int32 (JAX default int)
    float*       out = (float*)d_out;           // [B, C, TARGET] f32

    // Workspace layout: counts (u32) then reciprocals (f32).
    unsigned* cnt = (unsigned*)d_ws;
    float*    inv = (float*)((char*)d_ws + (size_t)B_ * TARGET_ * sizeof(unsigned));

    // 1) zero the output accumulator and the counts (every call: we accumulate).
    long long n4out = (long long)B_ * C_ * TARGET_ / 4;   // 15,360,000 float4
    mp_zero_f4<<<(unsigned)((n4out + 255) / 256), 256, 0, stream>>>((float4*)out, n4out);
    long long n4cnt = (long long)B_ * TARGET_ / 4;        // 60,000 uint4
    mp_zero_f4<<<(unsigned)((n4cnt + 255) / 256), 256, 0, stream>>>((float4*)cnt, n4cnt);

    // 2) occurrence histogram.
    mp_counts<<<(B_ * F_ + 255) / 256, 256, 0, stream>>>(gid, cnt);

    // 3) reciprocals of clamped counts.
    mp_recip<<<(B_ * TARGET_ + 255) / 256, 256, 0, stream>>>(cnt, inv);

    // 4) segmented-sum scatter (bandwidth-bound main pass).
    dim3 gs((F_ + FTILE - 1) / FTILE, C_ / CHSLAB, B_);   // 118 x 4 x 16 blocks
    mp_scatter<<<gs, FTILE, 0, stream>>>(fe, gid, out);

    // 5) turn sums into means.
    dim3 gd((TARGET_ + 255) / 256, C_, B_);               // 59 x 256 x 16 blocks
    mp_scale<<<gd, 256, 0, stream>>>(out, inv);
}